// ResGCNLayer_20547123544256
// MI455X (gfx1250) — compile-verified
//
#include <hip/hip_runtime.h>

#define NN 100000
#define DD 128
#define EE 1600000

typedef float v2f __attribute__((ext_vector_type(2)));
typedef float v8f __attribute__((ext_vector_type(8)));

// -----------------------------------------------------------------------------
// Kernel 1: h = x @ W^T + b  via V_WMMA_F32_16X16X4_F32, and out = x (residual).
// Block = 256 threads = 8 waves. Each block: 16 rows (M-tile), wave w handles
// output columns [16w, 16w+16). K-loop: 32 steps of K=4.
// -----------------------------------------------------------------------------
__global__ __launch_bounds__(256) void resgcn_gemm_kernel(
    const float* __restrict__ x, const float* __restrict__ W,
    const float* __restrict__ bias, float* __restrict__ h,
    float* __restrict__ out, int n_nodes)
{
    const int tid  = threadIdx.x;
    const int wave = tid >> 5;
    const int lane = tid & 31;
    const int m0   = blockIdx.x * 16;
    if (m0 + 16 > n_nodes) return;          // uniform guard; EXEC stays full
    const int n0   = wave * 16;

    // Residual init: out[m0:m0+16, :] = x[m0:m0+16, :]  (fully coalesced)
    {
        const size_t base = (size_t)m0 * DD;
        #pragma unroll
        for (int i = 0; i < 8; ++i) {
            const int off = i * 256 + tid;
            out[base + off] = x[base + off];
        }
    }

    // WMMA f32 16x16x4 fragment addressing (CDNA5 ISA layouts):
    //  A 16x4:  lanes 0-15 -> M=lane, VGPR0/1 = K=0/1 ; lanes 16-31 -> K=2/3
    //  B 4x16:  lanes 0-15 -> N=lane, VGPR0/1 = K=0/1 ; lanes 16-31 -> K=2/3
    //  B[k][n] = W[n][k]  (we need x @ W^T)
    const int l15   = lane & 15;
    const int khalf = (lane >> 4) << 1;     // 0 or 2

    const float* __restrict__ xrow = x + (size_t)(m0 + l15) * DD + khalf;
    const float* __restrict__ wrow = W + (size_t)(n0 + l15) * DD + khalf;

    v8f c = {};
    #pragma unroll 8
    for (int k = 0; k < DD; k += 4) {
        v2f a, b;
        a.x = xrow[k];  a.y = xrow[k + 1];
        b.x = wrow[k];  b.y = wrow[k + 1];
        // 8 args: (neg_a, A, neg_b, B, c_mod, C, reuse_a, reuse_b)
        c = __builtin_amdgcn_wmma_f32_16x16x4_f32(false, a, false, b,
                                                  (short)0, c, false, false);
    }

    // C/D layout: lane gives N = n0 + (lane&15); VGPR v -> M = v + 8*(lane>=16)
    const int   n     = n0 + l15;
    const float bn    = bias[n];
    const int   mbase = m0 + ((lane >> 4) << 3);
    #pragma unroll
    for (int v = 0; v < 8; ++v) {
        h[(size_t)(mbase + v) * DD + n] = c[v] + bn;
    }
}

// -----------------------------------------------------------------------------
// Kernel 2: scatter-add. One wave per edge (4 edges per wave), each lane owns a
// float4 slice of the 128-wide row. h and out rows are L2-resident (51 MB each
// < 192 MB L2), so gather + f32 atomics run at L2 speed.
// -----------------------------------------------------------------------------
__global__ __launch_bounds__(256) void resgcn_edge_kernel(
    const int* __restrict__ edge_row, const int* __restrict__ edge_col,
    const float* __restrict__ edge_weight, const float* __restrict__ h,
    float* __restrict__ agg, int n_edges)
{
    const int wave = threadIdx.x >> 5;
    const int lane = threadIdx.x & 31;
    const int e0   = (blockIdx.x * 8 + wave) * 4;

    #pragma unroll
    for (int i = 0; i < 4; ++i) {
        const int e = e0 + i;
        if (e >= n_edges) return;           // wave-uniform
        const int   row = edge_row[e];
        const int   col = edge_col[e];
        const float w   = edge_weight[e];

        const float4 v = *(const float4*)(h + (size_t)col * DD + lane * 4);
        float* dst = agg + (size_t)row * DD + lane * 4;
        unsafeAtomicAdd(dst + 0, v.x * w);   // -> global_atomic_add_f32
        unsafeAtomicAdd(dst + 1, v.y * w);
        unsafeAtomicAdd(dst + 2, v.z * w);
        unsafeAtomicAdd(dst + 3, v.w * w);
    }
}

// -----------------------------------------------------------------------------
// Kernel 3: in-place leaky_relu(out), slope 0.2, float4 per thread.
// -----------------------------------------------------------------------------
__global__ __launch_bounds__(256) void resgcn_lrelu_kernel(
    float* __restrict__ out, int total4)
{
    const int i = blockIdx.x * 256 + threadIdx.x;
    if (i >= total4) return;
    float4 v = *((float4*)out + i);
    v.x = v.x > 0.0f ? v.x : 0.2f * v.x;
    v.y = v.y > 0.0f ? v.y : 0.2f * v.y;
    v.z = v.z > 0.0f ? v.z : 0.2f * v.z;
    v.w = v.w > 0.0f ? v.w : 0.2f * v.w;
    *((float4*)out + i) = v;
}

extern "C" void kernel_launch(void* const* d_in, const int* in_sizes, int n_in,
                              void* d_out, int out_size, void* d_ws, size_t ws_size,
                              hipStream_t stream) {
    const float* x    = (const float*)d_in[0];
    const int*   erow = (const int*)  d_in[1];
    const int*   ecol = (const int*)  d_in[2];
    const float* ew   = (const float*)d_in[3];
    const float* W    = (const float*)d_in[4];
    const float* bias = (const float*)d_in[5];

    const int n_nodes = in_sizes[0] / DD;   // 100000
    const int n_edges = in_sizes[1];        // 1600000

    float* out = (float*)d_out;
    float* h   = (float*)d_ws;              // N*D floats = 51.2 MB scratch

    // 1) GEMM h = x@W^T + b ; out = x
    const int gemm_blocks = (n_nodes + 15) / 16;           // 6250
    resgcn_gemm_kernel<<<gemm_blocks, 256, 0, stream>>>(x, W, bias, h, out, n_nodes);

    // 2) out += segment_sum(h[col] * w)
    const int edge_blocks = (n_edges + 31) / 32;           // 50000 (8 waves x 4 edges)
    resgcn_edge_kernel<<<edge_blocks, 256, 0, stream>>>(erow, ecol, ew, h, out, n_edges);

    // 3) out = leaky_relu(out)
    const int total4 = (n_nodes * DD) / 4;                 // 3.2M float4
    resgcn_lrelu_kernel<<<(total4 + 255) / 256, 256, 0, stream>>>(out, total4);
}